// TripletLoss_12120397710007
// MI455X (gfx1250) — compile-verified
//
#include <hip/hip_runtime.h>

typedef float v2f __attribute__((ext_vector_type(2)));
typedef float v8f __attribute__((ext_vector_type(8)));

#define BN 8192
#define DK 128
#define MARGIN_F 0.3f
#define COL_SLICES 8
#define COLS_PER_SLICE (BN / COL_SLICES)   // 1024
#define ROWS_PER_BLOCK 64                  // 4 waves * 16 rows
#define POS_INF __builtin_inff()

// ---- monotone float <-> uint mapping (total order, handles +-inf) ----
__device__ __forceinline__ unsigned int fenc(float f) {
    unsigned int u = __float_as_uint(f);
    return (u & 0x80000000u) ? ~u : (u | 0x80000000u);
}
__device__ __forceinline__ float fdec(unsigned int u) {
    return __uint_as_float((u & 0x80000000u) ? (u ^ 0x80000000u) : ~u);
}

// ---------------- kernel 1: squared norms + accumulator init ----------------
// one wave per row: 32 lanes x float4 = 128 floats exactly
__global__ void __launch_bounds__(256)
tl_norms_init(const float* __restrict__ X, float* __restrict__ sqn,
              unsigned int* __restrict__ hp_enc, unsigned int* __restrict__ hn_enc) {
    int row  = (blockIdx.x * blockDim.x + threadIdx.x) >> 5;
    int lane = threadIdx.x & 31;
    const float4 v = ((const float4*)(X + (size_t)row * DK))[lane];
    float s = v.x * v.x + v.y * v.y + v.z * v.z + v.w * v.w;
#pragma unroll
    for (int m = 16; m >= 1; m >>= 1) s += __shfl_xor(s, m, 32);
    if (lane == 0) {
        sqn[row]    = s;
        hp_enc[row] = fenc(-POS_INF);   // running max of t over positives
        hn_enc[row] = fenc(POS_INF);    // running min of t over negatives
    }
}

// ---------------- kernel 2: fused Gram(WMMA f32) + batch-hard mining --------
// wave w of block handles rows [bx*64 + w*16, +16); block col-slice = by.
// C tile layout (ISA 7.12.2): lane l, vgpr r -> M = r + 8*(l>=16), N = l%16.
__global__ void __launch_bounds__(128)
tl_hard_mine(const float* __restrict__ X, const int* __restrict__ labels,
             const float* __restrict__ sqn,
             unsigned int* __restrict__ hp_enc, unsigned int* __restrict__ hn_enc) {
    const int lane = threadIdx.x & 31;
    const int wave = threadIdx.x >> 5;         // 0..3
    const int hi   = lane >> 4;                // 0 | 1
    const int lm   = lane & 15;

    const int rbase  = blockIdx.x * ROWS_PER_BLOCK + wave * 16;
    const int cbase0 = blockIdx.y * COLS_PER_SLICE;

    // A fragments for this wave's 16 rows, all of K, register-resident.
    // f32 A 16x4 layout: lane holds row M=l%16, K = 2*(l/16) + {0,1} -> 8B load
    v2f a[DK / 4];
    const float* arow = X + (size_t)(rbase + lm) * DK + hi * 2;
#pragma unroll
    for (int kk = 0; kk < DK / 4; ++kk) a[kk] = *(const v2f*)(arow + kk * 4);

    // row labels / indices covered by this lane (8 rows)
    int rl[8], ri[8];
#pragma unroll
    for (int r = 0; r < 8; ++r) {
        ri[r] = rbase + hi * 8 + r;
        rl[r] = labels[ri[r]];
    }

    float accp[8], accn[8];
#pragma unroll
    for (int r = 0; r < 8; ++r) { accp[r] = -POS_INF; accn[r] = POS_INF; }

    for (int ct = 0; ct < COLS_PER_SLICE / 16; ++ct) {
        const int cb = cbase0 + ct * 16;
        const float* brow = X + (size_t)(cb + lm) * DK + hi * 2;
        __builtin_prefetch(brow + 16 * DK, 0, 0);   // next col tile -> global_prefetch

        v8f c = {};
#pragma unroll
        for (int kk = 0; kk < DK / 4; ++kk) {
            const v2f b = *(const v2f*)(brow + kk * 4);
            c = __builtin_amdgcn_wmma_f32_16x16x4_f32(
                    false, a[kk], false, b, (short)0, c, false, false);
        }

        const int   col = cb + lm;         // this lane's column N
        const int   cl  = labels[col];
        const float sj  = sqn[col];
#pragma unroll
        for (int r = 0; r < 8; ++r) {
            const float t    = fmaf(-2.0f, c[r], sj);   // sq_j - 2*g  (monotone in dist)
            const bool  same = (rl[r] == cl);
            if (same && (ri[r] != col)) accp[r] = fmaxf(accp[r], t);
            if (!same)                  accn[r] = fminf(accn[r], t);
        }
    }

    // cross-lane reduce over the 16 lanes sharing each row group (masks <=8
    // never cross the 16-lane boundary), then one lane commits via atomics.
#pragma unroll
    for (int r = 0; r < 8; ++r) {
#pragma unroll
        for (int m = 8; m >= 1; m >>= 1) {
            accp[r] = fmaxf(accp[r], __shfl_xor(accp[r], m, 32));
            accn[r] = fminf(accn[r], __shfl_xor(accn[r], m, 32));
        }
        if (lm == 0) {   // lanes 0 and 16: rows rbase+r and rbase+8+r
            atomicMax(&hp_enc[ri[r]], fenc(accp[r]));
            atomicMin(&hn_enc[ri[r]], fenc(accn[r]));
        }
    }
}

// ---------------- kernel 3: finalize (sqrt, relu, mean) ---------------------
__global__ void __launch_bounds__(256)
tl_finalize(const float* __restrict__ sqn, const unsigned int* __restrict__ hp_enc,
            const unsigned int* __restrict__ hn_enc, float* __restrict__ out) {
    __shared__ float sm[256];
    float sum = 0.0f;
    for (int i = threadIdx.x; i < BN; i += 256) {
        const float tp = fdec(hp_enc[i]);
        const float tn = fdec(hn_enc[i]);
        const float si = sqn[i];
        // -inf (no positive) / +inf (no negative) propagate -> relu gives 0
        const float hp = (tp == -POS_INF) ? -POS_INF : sqrtf(fmaxf(tp + si, 0.0f));
        const float hn = (tn ==  POS_INF) ?  POS_INF : sqrtf(fmaxf(tn + si, 0.0f));
        sum += fmaxf(hp - hn + MARGIN_F, 0.0f);
    }
    sm[threadIdx.x] = sum;
    __syncthreads();
    for (int s = 128; s >= 1; s >>= 1) {
        if ((int)threadIdx.x < s) sm[threadIdx.x] += sm[threadIdx.x + s];
        __syncthreads();
    }
    if (threadIdx.x == 0) out[0] = sm[0] / (float)BN;
}

extern "C" void kernel_launch(void* const* d_in, const int* in_sizes, int n_in,
                              void* d_out, int out_size, void* d_ws, size_t ws_size,
                              hipStream_t stream) {
    const float* X      = (const float*)d_in[0];   // (8192,128) fp32
    const int*   labels = (const int*)d_in[1];     // (8192,) int32
    float* out = (float*)d_out;

    // workspace layout: sqn | hp_enc | hn_enc  (3 * 8192 * 4B = 96 KB)
    float*        sqn    = (float*)d_ws;
    unsigned int* hp_enc = (unsigned int*)(sqn + BN);
    unsigned int* hn_enc = hp_enc + BN;

    tl_norms_init<<<BN * 32 / 256, 256, 0, stream>>>(X, sqn, hp_enc, hn_enc);

    dim3 grid(BN / ROWS_PER_BLOCK, COL_SLICES);    // (128, 8) blocks, 4 waves each
    tl_hard_mine<<<grid, 128, 0, stream>>>(X, labels, sqn, hp_enc, hn_enc);

    tl_finalize<<<1, 256, 0, stream>>>(sqn, hp_enc, hn_enc, out);

    (void)in_sizes; (void)n_in; (void)out_size; (void)ws_size;
}